// single_Net_7876970021055
// MI455X (gfx1250) — compile-verified
//
#include <hip/hip_runtime.h>
#include <hip/hip_bf16.h>

#define N_NODES 50000
#define N_EDGES 1600000
#define IN_CH   512
#define OUT_CH  48
#define LDS_STRIDE 516   // 512 + 4 pad: row stride %64 == 4 banks -> spreads rows

typedef __attribute__((ext_vector_type(2))) float v2f;
typedef __attribute__((ext_vector_type(8))) float v8f;

// ---------------------------------------------------------------------------
// Kernel 1: XW = features[50000x512] @ weight[512x48], exact fp32 via
// V_WMMA_F32_16X16X4_F32. One block = 16 output rows, 3 waves = 3x16 cols.
// ---------------------------------------------------------------------------
__global__ __launch_bounds__(96) void gemm_xw(const float* __restrict__ feat,
                                              const float* __restrict__ w,
                                              float* __restrict__ xw) {
  __shared__ __align__(16) float As[16 * LDS_STRIDE];
  const int tid = threadIdx.x;
  const int m0  = blockIdx.x * 16;

  // Cooperative, coalesced float4 staging of the 16x512 A tile into LDS.
  for (int idx = tid; idx < 16 * (IN_CH / 4); idx += 96) {
    const int r  = idx >> 7;            // / (512/4)
    const int c4 = (idx & 127) << 2;    // float column
    float4 v = *(const float4*)(feat + (size_t)(m0 + r) * IN_CH + c4);
    *(float4*)(&As[r * LDS_STRIDE + c4]) = v;
  }
  __syncthreads();

  const int lane = tid & 31;
  const int n0   = (tid >> 5) * 16;     // wave -> 16-col tile (0,16,32)
  const int r    = lane & 15;           // M (for A) / N (for B)
  const int kh   = (lane >> 4) << 1;    // lane half selects K {0,1} vs {2,3}

  v8f acc = {};
  #pragma unroll 8
  for (int k = 0; k < IN_CH; k += 4) {
    // A 16x4 fp32: VGPR0 = K=kh, VGPR1 = K=kh+1 (ISA 7.12.2)
    v2f a, b;
    const float* ap = &As[r * LDS_STRIDE + k + kh];
    a.x = ap[0];
    a.y = ap[1];
    // B 4x16 fp32, row-major weight [512x48]; symmetric K striping by lane half
    const float* bp = w + (size_t)(k + kh) * OUT_CH + n0 + r;
    b.x = bp[0];
    b.y = bp[OUT_CH];
    acc = __builtin_amdgcn_wmma_f32_16x16x4_f32(
        /*neg_a=*/false, a, /*neg_b=*/false, b,
        /*c_mod=*/(short)0, acc, /*reuse_a=*/false, /*reuse_b=*/false);
  }

  // C/D layout: VGPR i -> row (m0 + i) for lanes 0-15, (m0 + 8 + i) for 16-31.
  const int col     = n0 + r;
  const int rowbase = m0 + ((lane >> 4) << 3);
  #pragma unroll
  for (int i = 0; i < 8; ++i)
    xw[(size_t)(rowbase + i) * OUT_CH + col] = acc[i];
}

// ---------------------------------------------------------------------------
// Kernel 2: scatter-add  aggr[dst] += w_e * XW[src].  12 threads per edge,
// 4 channels each: one float4 L2 gather + 4 native global_atomic_add_f32.
// ---------------------------------------------------------------------------
__global__ __launch_bounds__(256) void edge_scatter(const int* __restrict__ ei,
                                                    const float* __restrict__ ew,
                                                    const float* __restrict__ xw,
                                                    float* __restrict__ aggr) {
  const int gid = blockIdx.x * 256 + threadIdx.x;   // < 19.2M
  const int e   = gid / 12;
  const int c4  = (gid - e * 12) << 2;              // channel group * 4
  const int src = ei[e];
  const int dst = ei[N_EDGES + e];
  const float wgt = ew[e];

  float4 x = *(const float4*)(xw + (size_t)src * OUT_CH + c4);
  const float v0 = wgt * x.x, v1 = wgt * x.y, v2 = wgt * x.z, v3 = wgt * x.w;
  float* p = aggr + (size_t)dst * OUT_CH + c4;

  // Native L2 fp32 atomics (no CAS loop); one 64-bit addr pair, imm offsets.
  asm volatile(
      "global_atomic_add_f32 %0, %1, off\n\t"
      "global_atomic_add_f32 %0, %2, off offset:4\n\t"
      "global_atomic_add_f32 %0, %3, off offset:8\n\t"
      "global_atomic_add_f32 %0, %4, off offset:12"
      :
      : "v"(p), "v"(v0), "v"(v1), "v"(v2), "v"(v3)
      : "memory");
}

// ---------------------------------------------------------------------------
// Kernel 3: in-place z = aggr + bias; out = log_softmax(z) per node row (48).
// One wave32 per node: lane L holds ch L, lanes 0-15 also hold ch 32+L.
// ---------------------------------------------------------------------------
__global__ __launch_bounds__(256) void finalize_lsm(float* __restrict__ out,
                                                    const float* __restrict__ bias) {
  const int node = (blockIdx.x * 256 + threadIdx.x) >> 5;
  const int lane = threadIdx.x & 31;
  if (node >= N_NODES) return;
  float* row = out + (size_t)node * OUT_CH;

  float z0 = row[lane] + bias[lane];
  float z1 = (lane < 16) ? (row[32 + lane] + bias[32 + lane]) : -3.402823e38f;

  float m = fmaxf(z0, z1);
  #pragma unroll
  for (int off = 16; off; off >>= 1) m = fmaxf(m, __shfl_xor(m, off, 32));

  float s = __expf(z0 - m) + ((lane < 16) ? __expf(z1 - m) : 0.0f);
  #pragma unroll
  for (int off = 16; off; off >>= 1) s += __shfl_xor(s, off, 32);

  const float lse = m + __logf(s);
  row[lane] = z0 - lse;
  if (lane < 16) row[32 + lane] = z1 - lse;
}

// ---------------------------------------------------------------------------
extern "C" void kernel_launch(void* const* d_in, const int* in_sizes, int n_in,
                              void* d_out, int out_size, void* d_ws, size_t ws_size,
                              hipStream_t stream) {
  const int*   ei   = (const int*)d_in[0];    // edge_index [2, E]
  const float* feat = (const float*)d_in[1];  // [N, 512]
  const float* ew   = (const float*)d_in[2];  // [E]
  const float* w    = (const float*)d_in[3];  // [512, 48]
  const float* bias = (const float*)d_in[4];  // [48]
  float* out = (float*)d_out;                 // [N, 48]
  float* xw  = (float*)d_ws;                  // [N, 48] scratch

  // Zero the atomic accumulator (capture-legal memset node).
  hipMemsetAsync(out, 0, (size_t)N_NODES * OUT_CH * sizeof(float), stream);

  gemm_xw<<<N_NODES / 16, 96, 0, stream>>>(feat, w, xw);
  edge_scatter<<<(N_EDGES * 12) / 256, 256, 0, stream>>>(ei, ew, xw, out);
  finalize_lsm<<<N_NODES / 8, 256, 0, stream>>>(out, bias);
}